// MultiEmbeddingSGNS_37194416783620
// MI455X (gfx1250) — compile-verified
//
#include <hip/hip_runtime.h>

typedef __attribute__((ext_vector_type(2))) float v2f;
typedef __attribute__((ext_vector_type(8))) float v8f;

#define EMB_D 128
#define RSTR  132               // padded LDS row stride (floats) -> conflict-free ds_load_b64
#define SPW   16                // samples per wave (one WMMA tile)
#define WPB   2                 // waves per block
#define TPB   (WPB * 32)

__global__ __launch_bounds__(TPB) void sgns_wmma_kernel(
    const int*   __restrict__ x,
    const float* __restrict__ tgt_base, const float* __restrict__ tgt_a, const float* __restrict__ tgt_b,
    const float* __restrict__ ctx_base, const float* __restrict__ ctx_a, const float* __restrict__ ctx_b,
    const float* __restrict__ tgt_w,    const float* __restrict__ ctx_w,
    float* __restrict__ out)
{
    __shared__ float sh[WPB * 2 * SPW * RSTR];   // per wave: tv[16][132], cv[16][132]

    const int lane = threadIdx.x & 31;
    const int wave = threadIdx.x >> 5;
    const int s0   = (blockIdx.x * WPB + wave) * SPW;   // first sample of this wave

    float* tvb = sh + wave * (2 * SPW * RSTR);
    float* cvb = tvb + SPW * RSTR;

    // ---- softmax of the two 3-element weight vectors (in registers) ----
    float tw0 = tgt_w[0], tw1 = tgt_w[1], tw2 = tgt_w[2];
    float mt  = fmaxf(tw0, fmaxf(tw1, tw2));
    float te0 = __expf(tw0 - mt), te1 = __expf(tw1 - mt), te2 = __expf(tw2 - mt);
    float trs = 1.0f / (te0 + te1 + te2);
    tw0 = te0 * trs; tw1 = te1 * trs; tw2 = te2 * trs;

    float cw0 = ctx_w[0], cw1 = ctx_w[1], cw2 = ctx_w[2];
    float mc  = fmaxf(cw0, fmaxf(cw1, cw2));
    float ce0 = __expf(cw0 - mc), ce1 = __expf(cw1 - mc), ce2 = __expf(cw2 - mc);
    float crs = 1.0f / (ce0 + ce1 + ce2);
    cw0 = ce0 * crs; cw1 = ce1 * crs; cw2 = ce2 * crs;

    // ---- per-lane index preload: lane (s & 15) owns sample s0+s's 3 indices ----
    const int myS = s0 + (lane & 15);
    int j0 = x[3 * myS + 0];
    int j1 = x[3 * myS + 1];
    int j2 = x[3 * myS + 2];

    // ---- Phase 1: coalesced gather + weighted blend -> LDS (tv, cv rows) ----
    const int col = lane * 4;   // this lane's 4 contiguous floats of the 128-dim row
    #pragma unroll 4
    for (int s = 0; s < SPW; ++s) {
        const int i0 = __shfl(j0, s, 32);
        const int i1 = __shfl(j1, s, 32);
        const int i2 = __shfl(j2, s, 32);

        const float4 t0 = *(const float4*)(tgt_base + (size_t)i0 * EMB_D + col);
        const float4 t1 = *(const float4*)(tgt_a    + (size_t)i1 * EMB_D + col);
        const float4 t2 = *(const float4*)(tgt_b    + (size_t)i2 * EMB_D + col);
        float4 tv;
        tv.x = tw0 * t0.x + tw1 * t1.x + tw2 * t2.x;
        tv.y = tw0 * t0.y + tw1 * t1.y + tw2 * t2.y;
        tv.z = tw0 * t0.z + tw1 * t1.z + tw2 * t2.z;
        tv.w = tw0 * t0.w + tw1 * t1.w + tw2 * t2.w;
        *(float4*)(tvb + s * RSTR + col) = tv;   // row base s*528B is 16B aligned

        const float4 c0 = *(const float4*)(ctx_base + (size_t)i0 * EMB_D + col);
        const float4 c1 = *(const float4*)(ctx_a    + (size_t)i1 * EMB_D + col);
        const float4 c2 = *(const float4*)(ctx_b    + (size_t)i2 * EMB_D + col);
        float4 cv;
        cv.x = cw0 * c0.x + cw1 * c1.x + cw2 * c2.x;
        cv.y = cw0 * c0.y + cw1 * c1.y + cw2 * c2.y;
        cv.z = cw0 * c0.z + cw1 * c1.z + cw2 * c2.z;
        cv.w = cw0 * c0.w + cw1 * c1.w + cw2 * c2.w;
        *(float4*)(cvb + s * RSTR + col) = cv;
    }
    __syncthreads();

    // ---- Phase 2: D(16x16) = TV(16x128) x CV^T(128x16) via v_wmma_f32_16x16x4_f32 ----
    // A 16x4 f32 layout: lanes 0-15 hold (K=0,K=1), lanes 16-31 hold (K=2,K=3), M = lane&15.
    // B  4x16 layout: VGPR j holds rows {j, j+2} across lane halves, N = lane&15.
    // For both, this lane's fragment = 2 consecutive floats: row[lane&15][k + 2*(lane>>4)].
    const int m  = lane & 15;
    const int hi = lane >> 4;
    const float* arow = tvb + m * RSTR + 2 * hi;
    const float* brow = cvb + m * RSTR + 2 * hi;

    v8f acc = {};
    #pragma unroll
    for (int k = 0; k < EMB_D; k += 4) {
        v2f a = *(const v2f*)(arow + k);   // 8B aligned (528*s + 4k*4 + 8*hi)
        v2f b = *(const v2f*)(brow + k);
        acc = __builtin_amdgcn_wmma_f32_16x16x4_f32(
            /*neg_a=*/false, a, /*neg_b=*/false, b,
            /*c_mod=*/(short)0, acc, /*reuse_a=*/false, /*reuse_b=*/false);
    }

    // ---- diagonal extract + sigmoid ----
    // C/D layout: VGPR r -> (M=r, N=lane) on lanes 0-15, (M=8+r, N=lane-16) on lanes 16-31.
    // So D[r][r] sits in acc[r] at lane r; D[8+r][8+r] sits in acc[r] at lane 24+r.
    #pragma unroll
    for (int r = 0; r < 8; ++r) {
        const int srow = (lane == r) ? r : ((lane == 24 + r) ? (8 + r) : -1);
        if (srow >= 0) {
            const float d = acc[r];
            out[s0 + srow] = 1.0f / (1.0f + __expf(-d));
        }
    }
}

extern "C" void kernel_launch(void* const* d_in, const int* in_sizes, int n_in,
                              void* d_out, int out_size, void* d_ws, size_t ws_size,
                              hipStream_t stream) {
    const int*   x   = (const int*)  d_in[0];
    const float* tb  = (const float*)d_in[1];
    const float* ta  = (const float*)d_in[2];
    const float* tbb = (const float*)d_in[3];
    const float* cb  = (const float*)d_in[4];
    const float* ca  = (const float*)d_in[5];
    const float* cbb = (const float*)d_in[6];
    const float* tw  = (const float*)d_in[7];
    const float* cw  = (const float*)d_in[8];
    float* out = (float*)d_out;

    const int samples_per_block = WPB * SPW;               // 32
    const int blocks = out_size / samples_per_block;       // 16384/32 = 512
    sgns_wmma_kernel<<<blocks, TPB, 0, stream>>>(x, tb, ta, tbb, cb, ca, cbb, tw, cw, out);
}